// HypAttnAgg_43611097924223
// MI455X (gfx1250) — compile-verified
//
#include <hip/hip_runtime.h>
#include <hip/hip_bf16.h>
#include <math.h>

typedef __attribute__((ext_vector_type(2))) float v2f;
typedef __attribute__((ext_vector_type(8))) float v8f;

#define HEADS 4
#define DIM 128
#define FD 32
#define ALPHA 0.2f
#define MIN_NORM 1e-15f
#define MAXNORM 0.996f   // (1 - PROJ_EPS)/sqrt(C)

// ---------------------------------------------------------------------------
// Kernel 0: zero the output (used as the 'weighted' accumulator) and rowsum.
// ---------------------------------------------------------------------------
__global__ void hyp_init_kernel(float* __restrict__ out, float* __restrict__ rowsum,
                                long long nOut, long long nRow) {
    long long i = (long long)blockIdx.x * 256 + threadIdx.x;
    if (i < nOut) {
        out[i] = 0.0f;
    } else if (i < nOut + nRow) {
        rowsum[i - nOut] = 0.0f;
    }
}

// ---------------------------------------------------------------------------
// Kernel 1: logmap0(x) fused with h = x_t @ W via V_WMMA_F32_16X16X4_F32.
// Block = 128 threads (4 waves); each wave computes a 16-node tile of h.
// Output columns ordered c = h*32 + f so W2[d][c] = W[h][d][f].
// ---------------------------------------------------------------------------
__global__ __launch_bounds__(128) void hyp_gemm_logmap_kernel(
        const float* __restrict__ x, const float* __restrict__ W,
        float* __restrict__ hbuf, int N) {
    __shared__ float xs[4][16 * 132];     // stride 132 -> conflict-free banks
    const int wv   = threadIdx.x >> 5;
    const int lane = threadIdx.x & 31;
    const int base = blockIdx.x * 64 + wv * 16;

    // ---- Phase A: logmap0 for 16 rows, staged into LDS ----
    for (int i = 0; i < 16; ++i) {
        int node = base + i;
        if (node >= N) node = N - 1;                 // clamp (stores guarded)
        const float4 v = *reinterpret_cast<const float4*>(x + (long long)node * DIM + lane * 4);
        float ss = v.x * v.x + v.y * v.y + v.z * v.z + v.w * v.w;
        #pragma unroll
        for (int off = 16; off > 0; off >>= 1) ss += __shfl_xor(ss, off, 32);
        float nrm = fmaxf(sqrtf(ss), MIN_NORM);
        float t   = fminf(nrm, 1.0f - 1e-7f);
        float art = 0.5f * (log1pf(t) - log1pf(-t)); // artanh
        float scl = art / nrm;
        float* row = &xs[wv][i * 132 + lane * 4];
        row[0] = v.x * scl; row[1] = v.y * scl; row[2] = v.z * scl; row[3] = v.w * scl;
    }
    __syncthreads();

    // ---- Phase B: 8 N-tiles x 32 K-steps of fp32 WMMA ----
    const int mrow  = lane & 15;          // M row (A) / N col (B,C)
    const int kh    = (lane >> 4) << 1;   // lanes 0-15: K{0,1}; 16-31: K{2,3}
    const float* arow = &xs[wv][mrow * 132];

    for (int t = 0; t < 8; ++t) {
        v8f acc = {};
        const int c = t * 16 + mrow;                       // output column
        const float* bbase = W + (c >> 5) * (DIM * FD) + (c & 31);
        #pragma unroll
        for (int k0 = 0; k0 < DIM; k0 += 4) {
            const int klo = k0 + kh;
            v2f a, b;
            a.x = arow[klo];
            a.y = arow[klo + 1];
            b.x = bbase[klo * FD];
            b.y = bbase[(klo + 1) * FD];
            acc = __builtin_amdgcn_wmma_f32_16x16x4_f32(
                    false, a, false, b, (short)0, acc, false, false);
        }
        // C/D layout: VGPR g holds row g (lanes 0-15) / g+8 (lanes 16-31)
        #pragma unroll
        for (int g = 0; g < 8; ++g) {
            int row  = g + ((lane >> 4) << 3);
            int node = base + row;
            if (node < N) hbuf[(long long)node * DIM + t * 16 + mrow] = acc[g];
        }
    }
}

// ---------------------------------------------------------------------------
// Kernel 2: per-(node,head) attention score halves s_src, s_dst.
// ---------------------------------------------------------------------------
__global__ void hyp_score_kernel(const float* __restrict__ hbuf,
                                 const float* __restrict__ a,
                                 float* __restrict__ ssrc, float* __restrict__ sdst,
                                 int N) {
    int tid = blockIdx.x * 256 + threadIdx.x;
    if (tid >= N * HEADS) return;
    int n = tid >> 2, h = tid & 3;
    const float* hr = hbuf + (long long)n * DIM + h * FD;
    const float* ah = a + h * (2 * FD);
    float s1 = 0.0f, s2 = 0.0f;
    #pragma unroll
    for (int f = 0; f < FD; ++f) {
        float v = hr[f];
        s1 += v * ah[f];
        s2 += v * ah[FD + f];
    }
    ssrc[tid] = s1;
    sdst[tid] = s2;
}

// ---------------------------------------------------------------------------
// Kernel 3: one wave per edge. score -> e = exp(-leaky_relu) -> atomic
// scatter of e*h_dst into out (weighted) and e into rowsum. All gathers
// are L2-resident (h + out + scores ~= 110 MB < 192 MB L2).
// ---------------------------------------------------------------------------
__global__ __launch_bounds__(256) void hyp_edge_kernel(
        const int* __restrict__ ei, const float* __restrict__ hbuf,
        const float* __restrict__ ssrc, const float* __restrict__ sdst,
        float* __restrict__ outw, float* __restrict__ rowsum, int E) {
    const int wv   = threadIdx.x >> 5;
    const int lane = threadIdx.x & 31;
    const int e    = blockIdx.x * 8 + wv;
    if (e >= E) return;

    const int src = ei[e];
    const int dst = ei[E + e];

    // lanes 0-3 hold the 4 heads' score halves; broadcast via shuffle
    float s1 = ssrc[src * HEADS + (lane & 3)];
    float s2 = sdst[dst * HEADS + (lane & 3)];
    float sc = s1 + s2;
    float lr = sc > 0.0f ? sc : ALPHA * sc;
    float ev = expf(-lr);

    if (lane < HEADS) atomicAdd(&rowsum[src * HEADS + lane], ev);

    const float* hd = hbuf + (long long)dst * DIM;
    float*       od = outw + (long long)src * DIM;
    #pragma unroll
    for (int h = 0; h < HEADS; ++h) {
        float eh = __shfl(ev, h, 32);
        atomicAdd(&od[h * FD + lane], eh * hd[h * FD + lane]);
    }
}

// ---------------------------------------------------------------------------
// Kernel 4: one wave per node. h_prime = elu(weighted/rowsum), then
// expmap0 + proj (scale = min(tanh(n), maxnorm)/n), in-place on d_out.
// ---------------------------------------------------------------------------
__global__ __launch_bounds__(256) void hyp_final_kernel(
        float* __restrict__ out, const float* __restrict__ rowsum, int N) {
    const int wv   = threadIdx.x >> 5;
    const int lane = threadIdx.x & 31;
    const int n    = blockIdx.x * 8 + wv;
    if (n >= N) return;

    float* row = out + (long long)n * DIM;
    float vals[4];
    float ss = 0.0f;
    #pragma unroll
    for (int j = 0; j < 4; ++j) {
        float w = row[j * FD + lane];
        float r = rowsum[n * HEADS + j];
        float v = w / r;
        v = v > 0.0f ? v : expm1f(v);    // elu (alpha=1)
        vals[j] = v;
        ss += v * v;
    }
    #pragma unroll
    for (int off = 16; off > 0; off >>= 1) ss += __shfl_xor(ss, off, 32);
    float nrm = fmaxf(sqrtf(ss), MIN_NORM);
    float tn  = tanhf(nrm);
    float scl = fminf(tn, MAXNORM) / nrm;
    #pragma unroll
    for (int j = 0; j < 4; ++j) row[j * FD + lane] = vals[j] * scl;
}

// ---------------------------------------------------------------------------
extern "C" void kernel_launch(void* const* d_in, const int* in_sizes, int n_in,
                              void* d_out, int out_size, void* d_ws, size_t ws_size,
                              hipStream_t stream) {
    const float* x  = (const float*)d_in[0];
    const int*   ei = (const int*)d_in[1];
    const float* W  = (const float*)d_in[2];
    const float* a  = (const float*)d_in[3];
    float* out = (float*)d_out;

    const int N = in_sizes[0] / DIM;
    const int E = in_sizes[1] / 2;

    // workspace layout: h (N*128) | s_src (N*4) | s_dst (N*4) | rowsum (N*4)
    float* hbuf   = (float*)d_ws;
    float* ssrc   = hbuf + (size_t)N * DIM;
    float* sdst   = ssrc + (size_t)N * HEADS;
    float* rowsum = sdst + (size_t)N * HEADS;

    const long long nOut = (long long)N * DIM;
    const long long nRow = (long long)N * HEADS;
    const long long tot  = nOut + nRow;

    hyp_init_kernel<<<(int)((tot + 255) / 256), 256, 0, stream>>>(out, rowsum, nOut, nRow);
    hyp_gemm_logmap_kernel<<<(N + 63) / 64, 128, 0, stream>>>(x, W, hbuf, N);
    hyp_score_kernel<<<(N * HEADS + 255) / 256, 256, 0, stream>>>(hbuf, a, ssrc, sdst, N);
    hyp_edge_kernel<<<(E + 7) / 8, 256, 0, stream>>>(ei, hbuf, ssrc, sdst, out, rowsum, E);
    hyp_final_kernel<<<(N + 7) / 8, 256, 0, stream>>>(out, rowsum, N);
}